// Adversarial_score_33114197852259
// MI455X (gfx1250) — compile-verified
//
#include <hip/hip_runtime.h>
#include <hip/hip_bf16.h>

#define BATCH    32
#define N_DIM    512
#define N_CLASS  100
#define BIG_NEG  1000000.0f

typedef __attribute__((ext_vector_type(2))) float v2f;
typedef __attribute__((ext_vector_type(8))) float v8f;

// One block per batch. 512 threads = 16 wave32.
// Thread (i = tid&127, slice = tid>>7) accumulates n in [slice*128, slice*128+128)
// for class i of the j=label column (the only un-penalized column of the 100x100
// verification grid -- see analysis). Cross-slice reduction via
// V_WMMA_F32_16X16X4_F32 with an all-ones B matrix.
__global__ __launch_bounds__(512)
void adv_score_kernel(const float* __restrict__ z,     // (B, 2, 512)
                      const float* __restrict__ w_u,   // (B, 512, 100)
                      const float* __restrict__ b_u,   // (B, 100)
                      const float* __restrict__ w_l,   // (B, 512, 100)
                      const float* __restrict__ b_l,   // (B, 100)
                      const float* __restrict__ y,     // (B, 100) one-hot
                      float* __restrict__ out)         // (B,)
{
    __shared__ float s_c[N_DIM];
    __shared__ float s_r[N_DIM];
    __shared__ float s_wl[N_DIM];
    __shared__ float s_part[4 * 128];
    __shared__ float s_sums[128];
    __shared__ float s_max[512];
    __shared__ int   s_label;

    const int b   = blockIdx.x;
    const int tid = threadIdx.x;

    // --- find one-hot label -------------------------------------------------
    if (tid < N_CLASS) {
        if (y[b * N_CLASS + tid] > 0.5f) s_label = tid;
    }
    __syncthreads();
    const int label = s_label;

    // --- stage center, radius, and the w_l[:, label] column into LDS --------
    {
        const float* zb = z + (size_t)b * 2 * N_DIM;
        const float* wl = w_l + (size_t)b * N_DIM * N_CLASS + label;
        for (int n = tid; n < N_DIM; n += 512) {
            float zmin = zb[n];
            float zmax = zb[N_DIM + n];
            s_c[n]  = 0.5f * (zmin + zmax);
            s_r[n]  = 0.5f * (zmax - zmin);
            s_wl[n] = wl[(size_t)n * N_CLASS];
        }
    }
    __syncthreads();

    // --- main streaming accumulation ---------------------------------------
    const int i     = tid & 127;
    const int slice = tid >> 7;          // 0..3
    float acc = 0.0f;
    if (i < N_CLASS) {
        const float* wu = w_u + (size_t)b * N_DIM * N_CLASS + i;
        const int n0 = slice * 128;
        #pragma unroll 8
        for (int k = 0; k < 128; ++k) {
            const int n = n0 + k;
            float d = wu[(size_t)n * N_CLASS] - s_wl[n];   // w_f[b,n,i,label]
            acc = fmaf(s_c[n], d, acc);                    // lin contribution
            acc = fmaf(s_r[n], __builtin_fabsf(d), acc);   // amp (|d| = src mod)
        }
    }
    s_part[slice * 128 + i] = (i < N_CLASS) ? acc : 0.0f;
    __syncthreads();

    // --- cross-slice (K=4) reduction with V_WMMA_F32_16X16X4_F32 ------------
    // A (16x4 f32): lanes 0-15 hold K=0,1 in VGPR0,1 for rows M=0..15;
    //               lanes 16-31 hold K=2,3.  B = ones, so D[m,*] = sum_k A[m,k].
    if (tid < 256) {                      // waves 0..7, EXEC all-ones per wave
        const int lane = tid & 31;
        const int wv   = tid >> 5;        // 0..7
        const int i0   = wv * 16;
        const int mi   = i0 + (lane & 15);
        const int kb   = (lane < 16) ? 0 : 2;
        v2f a;
        a.x = s_part[kb * 128 + mi];
        a.y = s_part[(kb + 1) * 128 + mi];
        v2f bb;
        bb.x = 1.0f;
        bb.y = 1.0f;
        v8f c = {};
        c = __builtin_amdgcn_wmma_f32_16x16x4_f32(
                /*neg_a=*/false, a, /*neg_b=*/false, bb,
                /*c_mod=*/(short)0, c, /*reuse_a=*/false, /*reuse_b=*/false);
        // D layout (16x16 f32): VGPR m, lanes 0-15 -> row M=m; lanes 16-31 -> M=m+8.
        if (lane == 0) {
            #pragma unroll
            for (int m = 0; m < 8; ++m) s_sums[i0 + m] = c[m];
        } else if (lane == 16) {
            #pragma unroll
            for (int m = 0; m < 8; ++m) s_sums[i0 + 8 + m] = c[m];
        }
    }
    __syncthreads();

    // --- bias, penalty, block max -------------------------------------------
    float score = -3.0e38f;
    if (tid < N_CLASS) {
        float v = s_sums[tid]
                + b_u[b * N_CLASS + tid]
                - b_l[b * N_CLASS + label];
        if (tid == label) v -= BIG_NEG;   // (label,label) entry carries -1e6
        score = v;
    }
    s_max[tid] = score;
    __syncthreads();
    for (int s = 256; s > 0; s >>= 1) {
        if (tid < s) s_max[tid] = fmaxf(s_max[tid], s_max[tid + s]);
        __syncthreads();
    }
    if (tid == 0) out[b] = s_max[0];
}

extern "C" void kernel_launch(void* const* d_in, const int* in_sizes, int n_in,
                              void* d_out, int out_size, void* d_ws, size_t ws_size,
                              hipStream_t stream) {
    // setup_inputs order: x(0), z_tensor(1), w_u(2), b_u(3), w_l(4), b_l(5), y_tensor(6)
    const float* z   = (const float*)d_in[1];
    const float* w_u = (const float*)d_in[2];
    const float* b_u = (const float*)d_in[3];
    const float* w_l = (const float*)d_in[4];
    const float* b_l = (const float*)d_in[5];
    const float* y   = (const float*)d_in[6];
    float* out = (float*)d_out;
    adv_score_kernel<<<BATCH, 512, 0, stream>>>(z, w_u, b_u, w_l, b_l, y, out);
}